// DiabaticReadout_27212912787469
// MI455X (gfx1250) — compile-verified
//
#include <hip/hip_runtime.h>
#include <math.h>

// Batched 2x2 symmetric eigensolve (closed form), memory-bound streaming map.
// Traffic: 20 B/sample * 10M = 200 MB -> ~8.6 us floor at 23.3 TB/s HBM.
// Strategy: B128 NT loads/stores, global_prefetch_b8 one stride ahead,
// grid-stride loop with 16K wave32s in flight.

typedef float v4f __attribute__((ext_vector_type(4)));

__global__ __launch_bounds__(256) void eig2x2_kernel(
    const float* __restrict__ d0,
    const float* __restrict__ d1,
    const float* __restrict__ lam,
    float* __restrict__ out,
    int n)
{
    const int nvec   = n >> 2;                       // vec4 work items
    const int tid    = blockIdx.x * blockDim.x + threadIdx.x;
    const int stride = gridDim.x * blockDim.x;

    const v4f* __restrict__ a4 = (const v4f*)d0;
    const v4f* __restrict__ b4 = (const v4f*)d1;
    const v4f* __restrict__ l4 = (const v4f*)lam;
    v4f* __restrict__ o4 = (v4f*)out;

    for (int i = tid; i < nvec; i += stride) {
        // Prefetch next grid-stride iteration (gfx1250 global_prefetch_b8).
        const int ip = i + stride;
        if (ip < nvec) {
            __builtin_prefetch((const void*)(a4 + ip), 0, 0);
            __builtin_prefetch((const void*)(b4 + ip), 0, 0);
            __builtin_prefetch((const void*)(l4 + ip), 0, 0);
        }

        // Non-temporal B128 loads: zero reuse, don't pollute the 192MB L2.
        v4f a = __builtin_nontemporal_load(a4 + i);
        v4f b = __builtin_nontemporal_load(b4 + i);
        v4f l = __builtin_nontemporal_load(l4 + i);

        v4f hs = 0.5f * (a + b);    // mean
        v4f hd = 0.5f * (a - b);    // half-difference
        v4f r;                      // discriminant sqrt(hd^2 + lam^2)
        r.x = sqrtf(fmaf(hd.x, hd.x, l.x * l.x));
        r.y = sqrtf(fmaf(hd.y, hd.y, l.y * l.y));
        r.z = sqrtf(fmaf(hd.z, hd.z, l.z * l.z));
        r.w = sqrtf(fmaf(hd.w, hd.w, l.w * l.w));

        v4f e0 = hs - r;            // lower state (ascending like eigh)
        v4f e1 = hs + r;            // upper state

        // [N,2] row-major: interleave (e0,e1) pairs -> two B128 stores.
        v4f o0 = {e0.x, e1.x, e0.y, e1.y};
        v4f o1 = {e0.z, e1.z, e0.w, e1.w};
        __builtin_nontemporal_store(o0, o4 + 2 * i);
        __builtin_nontemporal_store(o1, o4 + 2 * i + 1);
    }

    // Scalar tail for n % 4 (N=10M is divisible by 4; kept for generality).
    for (int i = (nvec << 2) + tid; i < n; i += stride) {
        float a = d0[i], b = d1[i], l = lam[i];
        float hs = 0.5f * (a + b);
        float hd = 0.5f * (a - b);
        float r  = sqrtf(fmaf(hd, hd, l * l));
        out[2 * i + 0] = hs - r;
        out[2 * i + 1] = hs + r;
    }
}

extern "C" void kernel_launch(void* const* d_in, const int* in_sizes, int n_in,
                              void* d_out, int out_size, void* d_ws, size_t ws_size,
                              hipStream_t stream) {
    const float* d0  = (const float*)d_in[0];
    const float* d1  = (const float*)d_in[1];
    const float* lam = (const float*)d_in[2];
    float* out = (float*)d_out;
    const int n = in_sizes[0];

    const int nvec = n >> 2;
    int blocks = (nvec + 255) / 256;
    if (blocks < 1)    blocks = 1;
    if (blocks > 2048) blocks = 2048;   // ~5 loop trips/thread -> prefetch has work to do

    eig2x2_kernel<<<blocks, 256, 0, stream>>>(d0, d1, lam, out, n);
}